// MOERouter_35845797053214
// MI455X (gfx1250) — compile-verified
//
#include <hip/hip_runtime.h>

typedef __attribute__((ext_vector_type(2))) float v2f;
typedef __attribute__((ext_vector_type(4))) float v4f;
typedef __attribute__((ext_vector_type(8))) float v8f;

#define B_   16
#define C_   64
#define H_   128
#define W_   128
#define HW_  (H_ * W_)          // 16384
#define NPIX (B_ * HW_)         // 262144
#define E_   64
#define K_   2

// ---------------------------------------------------------------------------
// Top-2 merge helper (tie-break: lower index wins, matching lax.top_k)
// ---------------------------------------------------------------------------
__device__ __forceinline__ void top2_merge(float v, int i,
                                           float& b0, int& b0i,
                                           float& b1, int& b1i) {
    if (v > b0 || (v == b0 && i < b0i)) {
        b1 = b0; b1i = b0i;
        b0 = v;  b0i = i;
    } else if (v > b1 || (v == b1 && i < b1i)) {
        b1 = v;  b1i = i;
    }
}

// ---------------------------------------------------------------------------
// Fused gate GEMM (WMMA f32 16x16x4) + bias + top-2 + renormalized weights.
// One wave handles 16 consecutive pixels (same b, same row since W%16==0)
// and all 64 experts (4 N-tiles of 16).
// ---------------------------------------------------------------------------
__global__ __launch_bounds__(256)
void moe_router_gemm_kernel(const float* __restrict__ x,       // [B,C,H,W]
                            const float* __restrict__ gw,      // [E,C]
                            const float* __restrict__ gb,      // [E]
                            float* __restrict__ logits,        // [B,E,H,W]
                            float* __restrict__ weights,       // [B,2,H,W]
                            float* __restrict__ indices)       // [B,2,H,W]
{
    const int lane = threadIdx.x & 31;
    const int wave = blockIdx.x * (blockDim.x >> 5) + (threadIdx.x >> 5);
    const int pixBase = wave * 16;          // 16 pixels per wave
    const int b  = pixBase / HW_;
    const int hw = pixBase % HW_;           // multiple of 16

    const int half = lane >> 4;             // 0 or 1
    const int l16  = lane & 15;

    // ---- Load A tiles: lane holds pixel l16; K = 4*kb + 2*half + r --------
    const float* xb = x + (size_t)b * C_ * HW_ + hw + l16;
    v2f a[16];
#pragma unroll
    for (int kb = 0; kb < 16; ++kb) {
        const int c0 = 4 * kb + 2 * half;
        a[kb].x = xb[(size_t)c0 * HW_];
        a[kb].y = xb[(size_t)(c0 + 1) * HW_];
    }

    // ---- GEMM: 4 expert tiles of 16, accumulate over K=64 -----------------
    v8f acc[4];
#pragma unroll
    for (int t = 0; t < 4; ++t) {
        const int e = 16 * t + l16;                 // this lane's expert (N)
        v8f c = {};
#pragma unroll
        for (int kb = 0; kb < 16; ++kb) {
            const int c0 = 4 * kb + 2 * half;
            v2f bv = *(const v2f*)(gw + e * C_ + c0);
            c = __builtin_amdgcn_wmma_f32_16x16x4_f32(
                    /*neg_a=*/false, a[kb],
                    /*neg_b=*/false, bv,
                    /*c_mod=*/(short)0, c,
                    /*reuse_a=*/false, /*reuse_b=*/false);
        }
        const float bias = gb[e];
#pragma unroll
        for (int r = 0; r < 8; ++r) c[r] += bias;
        acc[t] = c;
    }

    // ---- Store logits: lane L holds expert e, pixels (8*half + r), r=0..7 -
#pragma unroll
    for (int t = 0; t < 4; ++t) {
        float* lp = logits + ((size_t)b * E_ + (16 * t + l16)) * HW_
                           + hw + 8 * half;
        v4f lo = { acc[t][0], acc[t][1], acc[t][2], acc[t][3] };
        v4f hi = { acc[t][4], acc[t][5], acc[t][6], acc[t][7] };
        *(v4f*)lp       = lo;
        *((v4f*)lp + 1) = hi;
    }

    // ---- Top-2 over experts per pixel --------------------------------------
    // For fixed r and half, the 64 expert logits of pixel (8*half + r) live
    // in 4 tiles across the 16 lanes of this half-wave.
#pragma unroll
    for (int r = 0; r < 8; ++r) {
        float b0 = acc[0][r]; int b0i = l16;
        float b1 = -3.402823466e+38f; int b1i = 0x7fffffff;
        top2_merge(acc[1][r], 16 + l16, b0, b0i, b1, b1i);
        top2_merge(acc[2][r], 32 + l16, b0, b0i, b1, b1i);
        top2_merge(acc[3][r], 48 + l16, b0, b0i, b1, b1i);

#pragma unroll
        for (int m = 1; m < 16; m <<= 1) {
            float ob0 = __shfl_xor(b0, m, 16);
            int   oi0 = __shfl_xor(b0i, m, 16);
            float ob1 = __shfl_xor(b1, m, 16);
            int   oi1 = __shfl_xor(b1i, m, 16);
            top2_merge(ob0, oi0, b0, b0i, b1, b1i);
            top2_merge(ob1, oi1, b0, b0i, b1, b1i);
        }

        if (l16 == 0) {
            const int p = hw + 8 * half + r;
            const float e2 = __expf(b1 - b0);      // b1 <= b0, stable
            const float inv = 1.0f / (1.0f + e2);
            float* wp = weights + (size_t)b * K_ * HW_ + p;
            float* ip = indices + (size_t)b * K_ * HW_ + p;
            wp[0]   = inv;
            wp[HW_] = e2 * inv;
            ip[0]   = (float)b0i;
            ip[HW_] = (float)b1i;
        }
    }
}

// ---------------------------------------------------------------------------
// Expert-mask expansion: mask[e][k][n] = (indices[b(n)][k][hw(n)] == e)
// Output layout [E, K, B*H*W]; each thread writes 4 consecutive n (b128).
// ---------------------------------------------------------------------------
__global__ __launch_bounds__(256)
void moe_mask_kernel(const float* __restrict__ indices,  // [B,2,H,W]
                     float* __restrict__ mask)           // [E,2,NPIX]
{
    const size_t tid  = (size_t)blockIdx.x * blockDim.x + threadIdx.x;
    const size_t base = tid * 4;                 // flat index into [E,2,NPIX]
    const int n0 = (int)(base % NPIX);           // NPIX % 4 == 0 -> same (e,k)
    const int ek = (int)(base / NPIX);
    const int k  = ek & 1;
    const int e  = ek >> 1;
    const int bb = n0 / HW_;                     // HW_ % 4 == 0 -> same b
    const int hw = n0 % HW_;

    const float fe = (float)e;
    const float* ip = indices + ((size_t)bb * K_ + k) * HW_ + hw;
    v4f o;
    o.x = (ip[0] == fe) ? 1.0f : 0.0f;
    o.y = (ip[1] == fe) ? 1.0f : 0.0f;
    o.z = (ip[2] == fe) ? 1.0f : 0.0f;
    o.w = (ip[3] == fe) ? 1.0f : 0.0f;
    *(v4f*)(mask + base) = o;
}

// ---------------------------------------------------------------------------
extern "C" void kernel_launch(void* const* d_in, const int* in_sizes, int n_in,
                              void* d_out, int out_size, void* d_ws, size_t ws_size,
                              hipStream_t stream) {
    const float* x  = (const float*)d_in[0];   // [B,C,H,W]
    const float* gw = (const float*)d_in[1];   // [E,C]
    const float* gb = (const float*)d_in[2];   // [E]

    float* out     = (float*)d_out;
    float* logits  = out;                                     // B*E*H*W
    float* weights = logits  + (size_t)B_ * E_ * HW_;         // B*2*H*W
    float* indices = weights + (size_t)B_ * K_ * HW_;         // B*2*H*W
    float* mask    = indices + (size_t)B_ * K_ * HW_;         // E*2*NPIX

    // 16384 waves, 8 waves (256 threads) per block -> 2048 blocks
    const int waves  = NPIX / 16;
    const int blocks = waves / 8;
    moe_router_gemm_kernel<<<blocks, 256, 0, stream>>>(x, gw, gb,
                                                       logits, weights, indices);

    const size_t maskElems = (size_t)E_ * K_ * NPIX;          // 33,554,432
    const int mblocks = (int)(maskElems / 4 / 256);           // 32768
    moe_mask_kernel<<<mblocks, 256, 0, stream>>>(indices, mask);
}